// hadamard_transform_matmul_4904852652181
// MI455X (gfx1250) — compile-verified
//
#include <hip/hip_runtime.h>

// ---------------------------------------------------------------------------
// y = x @ H4096, H4096 = H16 (kron) H16 (kron) H16  (Sylvester Hadamard).
// 3-stage radix-16 fast Walsh-Hadamard transform per row; each radix-16
// butterfly batch is a 16x16x16 f32 matmul via four chained
// V_WMMA_F32_16X16X4_F32 (exact in fp32 since H entries are +/-1).
//
// MI455X roofline: 268 MB total I/O @ 23.3 TB/s ~= 11.5 us; compute is
// ~3.2 GFLOP total -> purely HBM-bound. Row copies global<->LDS attempt the
// Tensor Data Mover (one descriptor per 16KB row); supports both the
// ROCm 7.2 5-arg builtin and the therock 6-arg builtin (header-detected).
// Each wave32 owns one private row in LDS -> no barriers needed at all.
// ---------------------------------------------------------------------------

typedef float v2f __attribute__((ext_vector_type(2)));
typedef float v8f __attribute__((ext_vector_type(8)));
typedef unsigned int u32x4 __attribute__((ext_vector_type(4)));
typedef int i32x4 __attribute__((ext_vector_type(4)));
typedef int i32x8 __attribute__((ext_vector_type(8)));

#define DIM 4096
#define WAVES_PER_BLOCK 4
#define BLOCK_THREADS (WAVES_PER_BLOCK * 32)
#define ROW_BYTES (DIM * 4)

// ---- TDM capability probe (with stderr diagnostics) ----
#if !defined(__gfx1250__)
  #define TDM_ARGS 0   // host pass
#elif !__has_builtin(__builtin_amdgcn_tensor_load_to_lds) || \
      !__has_builtin(__builtin_amdgcn_tensor_store_from_lds) || \
      !__has_builtin(__builtin_amdgcn_s_wait_tensorcnt)
  #define TDM_ARGS 0
  #pragma message("gfx1250 device pass: TDM builtins NOT declared -> float4 fallback")
#elif __has_include(<hip/amd_detail/amd_gfx1250_TDM.h>)
  #define TDM_ARGS 6   // therock-10.0 headers present -> clang-23 6-arg builtin
  #pragma message("gfx1250 device pass: TDM via 6-arg builtin (TDM header present)")
#else
  #define TDM_ARGS 5   // ROCm 7.2 clang-22 5-arg builtin
  #pragma message("gfx1250 device pass: TDM via 5-arg builtin (no TDM header)")
#endif

#if TDM_ARGS > 0
// Build D# groups for a 1-D contiguous tile of DIM f32 elements.
// Field packing per CDNA5 ISA section 8.3/8.4:
//   g0: [1:0]=count=1, [63:32]=lds_addr, [120:64]=global_addr, [127:126]=type=2
//   g1: [17:16]=data_size=2 (4B), [79:48]=tensor_dim0, [127:112]=tile_dim0,
//       [143:128]=tile_dim1 (0 = unused -> 1-D), [207:160]=tensor_dim0_stride
__device__ __forceinline__ void tdm_row_desc(unsigned long long gaddr,
                                             unsigned lds_byte_off,
                                             u32x4& g0, i32x8& g1) {
    g0[0] = 1u;                                               // count=1, user D#
    g0[1] = lds_byte_off;                                     // lds_addr
    g0[2] = (unsigned)(gaddr & 0xFFFFFFFFull);                // global_addr[31:0]
    g0[3] = (unsigned)((gaddr >> 32) & 0x01FFFFFFull)         // global_addr[56:32]
          | (2u << 30);                                       // type = 2 ("image")
    g1[0] = (int)(2u << 16);                                  // data_size = 4 bytes
    g1[1] = (int)((DIM & 0xFFFF) << 16);                      // tensor_dim0[15:0]
    g1[2] = (int)((DIM >> 16) & 0xFFFF);                      // tensor_dim0[31:16]
    g1[3] = (int)((DIM & 0xFFFF) << 16);                      // tile_dim0 = DIM
    g1[4] = 0;                                                // tile_dim1/2 = 0 (1-D tile)
    g1[5] = DIM;                                              // tensor_dim0_stride[31:0]
    g1[6] = 0;
    g1[7] = 0;
}

__device__ __forceinline__ void tdm_load_row(unsigned long long gaddr,
                                             unsigned lds_byte_off) {
    u32x4 g0; i32x8 g1;
    tdm_row_desc(gaddr, lds_byte_off, g0, g1);
    i32x4 z4 = {0, 0, 0, 0};
#if TDM_ARGS == 5
    __builtin_amdgcn_tensor_load_to_lds(g0, g1, z4, z4, 0);
#else
    i32x8 z8 = {0, 0, 0, 0, 0, 0, 0, 0};
    __builtin_amdgcn_tensor_load_to_lds(g0, g1, z4, z4, z8, 0);
#endif
}

__device__ __forceinline__ void tdm_store_row(unsigned long long gaddr,
                                              unsigned lds_byte_off) {
    u32x4 g0; i32x8 g1;
    tdm_row_desc(gaddr, lds_byte_off, g0, g1);
    i32x4 z4 = {0, 0, 0, 0};
#if TDM_ARGS == 5
    __builtin_amdgcn_tensor_store_from_lds(g0, g1, z4, z4, 0);
#else
    i32x8 z8 = {0, 0, 0, 0, 0, 0, 0, 0};
    __builtin_amdgcn_tensor_store_from_lds(g0, g1, z4, z4, z8, 0);
#endif
}
#endif  // TDM_ARGS > 0

__global__ __launch_bounds__(BLOCK_THREADS)
void fwht4096_wmma_kernel(const float* __restrict__ x, float* __restrict__ y) {
    // One private row of 4096 floats per wave: 4 * 16KB = 64KB static LDS.
    __shared__ float lds[WAVES_PER_BLOCK][DIM];

    const int lane = threadIdx.x & 31;
    const int wave = threadIdx.x >> 5;
    const int row  = blockIdx.x * WAVES_PER_BLOCK + wave;

    const float* __restrict__ src = x + (size_t)row * DIM;
    float* __restrict__ dst       = y + (size_t)row * DIM;
    float* __restrict__ L         = lds[wave];

    // ---- Load row into LDS ----
#if TDM_ARGS > 0
    tdm_load_row((unsigned long long)(const void*)src,
                 (unsigned)(wave * ROW_BYTES));                 // TENSORcnt++
    __builtin_amdgcn_s_wait_tensorcnt(0);                       // row resident
    asm volatile("" ::: "memory");  // LDS written behind the compiler's back
#else
    {
        const float4* __restrict__ sv = (const float4*)src;
        float4* __restrict__ Lv = (float4*)L;
        #pragma unroll
        for (int i = 0; i < DIM / (32 * 4); ++i)
            Lv[i * 32 + lane] = sv[i * 32 + lane];
    }
#endif

    const int m    = lane & 15;   // M row of A/D tile, N column index for B/D
    const int half = lane >> 4;   // lane half selects K/M sub-block per ISA layout

    // H16 A-matrix chunks, hoisted: lane l, VGPR j holds A[m, 4c + j + 2*half]
    // with H16[m,k] = (-1)^popcount(m & k). 8 constants per lane, whole kernel.
    v2f A[4];
    #pragma unroll
    for (int c = 0; c < 4; ++c) {
        const int k0 = 4 * c + 2 * half;
        A[c].x = (__builtin_popcount(m & k0)       & 1) ? -1.0f : 1.0f;
        A[c].y = (__builtin_popcount(m & (k0 + 1)) & 1) ? -1.0f : 1.0f;
    }

    // Stage strides: element offset within the row = t*St + n*Sn + k*Sk
    //   stage 0 (axis stride 1):   (256, 16,   1)
    //   stage 1 (axis stride 16):  (256,  1,  16)
    //   stage 2 (axis stride 256): ( 16,  1, 256)
    #pragma unroll
    for (int s = 0; s < 3; ++s) {
        const int St = (s == 0) ? 256 : (s == 1) ? 256 : 16;
        const int Sn = (s == 0) ? 16  : 1;
        const int Sk = (s == 0) ? 1   : (s == 1) ? 16 : 256;

        for (int t = 0; t < 16; ++t) {
            const int base = t * St + m * Sn;   // this lane's column (n = m)

            v8f acc = {};
            #pragma unroll
            for (int c = 0; c < 4; ++c) {
                const int k0 = 4 * c + 2 * half;
                // B-matrix 4x16 chunk: lane holds B[K=j+2*half, N=m]
                v2f b;
                b.x = L[base + k0 * Sk];
                b.y = L[base + (k0 + 1) * Sk];
                // D = A*B + C  (neg_a, A, neg_b, B, c_mod, C, reuse_a, reuse_b)
                acc = __builtin_amdgcn_wmma_f32_16x16x4_f32(
                    false, A[c], false, b, (short)0, acc, false, false);
            }

            // D 16x16 f32 layout: VGPR r, lane l -> D[M = r + 8*half, N = m].
            // In-place writeback (tiles disjoint; same-wave LDS ops are ordered).
            #pragma unroll
            for (int r = 0; r < 8; ++r)
                L[base + (r + 8 * half) * Sk] = acc[r];
        }
    }

    // ---- Store row ----
#if TDM_ARGS > 0
    asm volatile("s_wait_dscnt 0x0" ::: "memory");  // LDS writes visible to TDM
    tdm_store_row((unsigned long long)(void*)dst,
                  (unsigned)(wave * ROW_BYTES));                // TENSORcnt++
    __builtin_amdgcn_s_wait_tensorcnt(0);  // (S_ENDPGM also implies wait-idle)
#else
    {
        float4* __restrict__ dv = (float4*)dst;
        const float4* __restrict__ Lv = (const float4*)L;
        #pragma unroll
        for (int i = 0; i < DIM / (32 * 4); ++i)
            dv[i * 32 + lane] = Lv[i * 32 + lane];
    }
#endif
}

extern "C" void kernel_launch(void* const* d_in, const int* in_sizes, int n_in,
                              void* d_out, int out_size, void* d_ws, size_t ws_size,
                              hipStream_t stream) {
    const float* x = (const float*)d_in[0];
    // d_in[1] (H) is deliberately unused: the Hadamard structure is synthesized
    // in-register from popcount parity (exact, zero memory traffic).
    float* out = (float*)d_out;

    const int n_tokens = in_sizes[0] / DIM;          // 8192
    const int n_blocks = n_tokens / WAVES_PER_BLOCK; // 2048

    fwht4096_wmma_kernel<<<n_blocks, BLOCK_THREADS, 0, stream>>>(x, out);
}